// FeatureRecalibration3D_4750233830124
// MI455X (gfx1250) — compile-verified
//
#include <hip/hip_runtime.h>
#include <math.h>

typedef __attribute__((ext_vector_type(2))) float v2f;
typedef __attribute__((ext_vector_type(8))) float v8f;

#define B_ 2
#define C_ 64
#define D_ 64
#define H_ 96
#define W_ 96
#define HW_ (H_ * W_)
#define DHW_ (D_ * H_ * W_)
#define HID_ 8
#define NEG_SLOPE 0.01f

#define SLABS 8
#define DPS (D_ / SLABS) // d-planes per block in pass 1

// workspace layout (floats)
#define PART_OFF 0      // B_*C_*SLABS = 1024 partial sums
#define WEFF_OFF 1024   // C_*27 effective boundary-class weights
#define GATE_OFF 2816   // B_*C_ sigmoid gates

// ---------------------------------------------------------------------------
// Pass 0: fold the 3x3x3 depthwise kernel into 27 boundary-class weights.
// wsum(class) = sum of taps k whose conv output window covers a position of
// that class: first index -> k<=1, last index -> k>=1, interior -> all.
// ---------------------------------------------------------------------------
__global__ void se3d_init_weff(const float* __restrict__ dw_w,
                               float* __restrict__ ws) {
  int c = threadIdx.x;
  if (c >= C_) return;
  const float* w = dw_w + c * 27;
  float* weff = ws + WEFF_OFF + c * 27;
  for (int cd = 0; cd < 3; ++cd)
    for (int ch = 0; ch < 3; ++ch)
      for (int cw = 0; cw < 3; ++cw) {
        float s = 0.f;
        for (int kd = 0; kd < 3; ++kd) {
          if (cd == 1 && kd > 1) continue;
          if (cd == 2 && kd < 1) continue;
          for (int kh = 0; kh < 3; ++kh) {
            if (ch == 1 && kh > 1) continue;
            if (ch == 2 && kh < 1) continue;
            for (int kw = 0; kw < 3; ++kw) {
              if (cw == 1 && kw > 1) continue;
              if (cw == 2 && kw < 1) continue;
              s += w[kd * 9 + kh * 3 + kw];
            }
          }
        }
        weff[cd * 9 + ch * 3 + cw] = s;
      }
}

// ---------------------------------------------------------------------------
// Pass 1: single streaming pass over x (adjoint trick: conv+mean collapses to
// one weighted accumulate per element, weight chosen by boundary class).
// Each block = one (b,c) channel and an 8-plane d-slab; float4 loads; block
// reduction -> one deterministic partial per block.
// ---------------------------------------------------------------------------
__global__ void se3d_conv_pool(const float* __restrict__ x,
                               float* __restrict__ ws) {
  __shared__ float red[256];
  const int blk = blockIdx.x;      // 0 .. B_*C_*SLABS-1
  const int bc = blk >> 3;
  const int slab = blk & 7;
  const float* weff = ws + WEFF_OFF + (bc & (C_ - 1)) * 27;

  float acc = 0.f;
  for (int dd = 0; dd < DPS; ++dd) {
    const int d = slab * DPS + dd;
    const int dcls = (d == 0) ? 1 : (d == D_ - 1) ? 2 : 0;
    float wr[9];
#pragma unroll
    for (int i = 0; i < 9; ++i) wr[i] = weff[dcls * 9 + i];

    const float4* xp = reinterpret_cast<const float4*>(
        x + ((size_t)bc * D_ + d) * (size_t)HW_);
    for (int t = threadIdx.x; t < HW_ / 4; t += 256) {
      const int h = t / (W_ / 4);
      const int w4 = t % (W_ / 4);
      const int hc = (h == 0) ? 1 : (h == H_ - 1) ? 2 : 0;
      const float4 v = xp[t];
      const float wfirst = wr[hc * 3 + ((w4 == 0) ? 1 : 0)];
      const float wmid = wr[hc * 3];
      const float wlast = wr[hc * 3 + ((w4 == W_ / 4 - 1) ? 2 : 0)];
      acc += v.x * wfirst + v.y * wmid + v.z * wmid + v.w * wlast;
    }
  }
  red[threadIdx.x] = acc;
  __syncthreads();
  for (int s = 128; s > 0; s >>= 1) {
    if (threadIdx.x < s) red[threadIdx.x] += red[threadIdx.x + s];
    __syncthreads();
  }
  if (threadIdx.x == 0) ws[PART_OFF + blk] = red[0];
}

// ---------------------------------------------------------------------------
// Pass 2: single wave32. Reduce partials -> z[B,C]; MLP via
// v_wmma_f32_16x16x4_f32.
//   GEMM1: h[2,8]  = z[2,64] @ fc1_w[8,64]^T   (16 K-steps of 4)
//   GEMM2: g[2,64] = h[2,8]  @ fc2_w[64,8]^T   (4 N-chunks x 2 K-steps)
// All WMMA operands come from zero-PADDED LDS tiles (or fully-valid global
// rows) so every operand load is an unconditional 64-bit load -- no EXEC
// masking in the WMMA pipeline.
// A layout: lane m=lane&15 holds row m, VGPRs {0,1} hold K = (lane>>4)*2+{0,1}.
// B layout: lane n=lane&15 holds col n, same K striping.
// C/D layout: element (M = r + (lane>>4)*8, N = lane&15) in VGPR r.
// ---------------------------------------------------------------------------
__global__ void se3d_mlp_wmma(const float* __restrict__ fc1_w,
                              const float* __restrict__ fc1_b,
                              const float* __restrict__ fc2_w,
                              const float* __restrict__ fc2_b,
                              float* __restrict__ ws) {
  __shared__ float zs[16 * C_];    // rows 0..1 = z, rows 2..15 = 0
  __shared__ float w1[16 * C_];    // rows 0..7 = fc1_w, rows 8..15 = 0
  __shared__ float hs[16 * HID_];  // rows 0..1 = h, rows 2..15 = 0
  const int lane = threadIdx.x;

  // partials -> z (mean over D*H*W), zero-pad rows >= B_
  for (int i = lane; i < 16 * C_; i += 32) {
    float v = 0.f;
    if (i < B_ * C_) {
      float s = 0.f;
#pragma unroll
      for (int p = 0; p < SLABS; ++p) s += ws[PART_OFF + i * SLABS + p];
      v = s * (1.f / (float)DHW_);
    }
    zs[i] = v;
  }
  // fc1_w -> LDS, zero-pad rows >= HID_
  for (int i = lane; i < 16 * C_; i += 32)
    w1[i] = (i < HID_ * C_) ? fc1_w[i] : 0.f;
  // zero h scratch (rows >= B_ stay zero)
  for (int i = lane; i < 16 * HID_; i += 32) hs[i] = 0.f;
  __syncthreads();

  const int m = lane & 15;
  const int grp = lane >> 4;

  // ---- GEMM1 ----
  v8f acc = {};
#pragma unroll
  for (int kb = 0; kb < 16; ++kb) {
    const int k0 = kb * 4 + grp * 2;
    const v2f a = *(const v2f*)&zs[m * C_ + k0];  // ds_load_b64, unconditional
    const v2f b = *(const v2f*)&w1[m * C_ + k0];  // ds_load_b64, unconditional
    acc = __builtin_amdgcn_wmma_f32_16x16x4_f32(false, a, false, b, (short)0,
                                                acc, false, false);
  }
  // bias + LeakyReLU, scatter h to LDS (cross-lane relayout for GEMM2 A)
#pragma unroll
  for (int r = 0; r < 8; ++r) {
    const int M = r + grp * 8;
    if (M < B_ && m < HID_) {
      float v = acc[r] + fc1_b[m];
      v = (v >= 0.f) ? v : NEG_SLOPE * v;
      hs[M * HID_ + m] = v;
    }
  }
  __syncthreads();

  // ---- GEMM2 ----
#pragma unroll
  for (int chunk = 0; chunk < 4; ++chunk) {
    v8f acc2 = {};
#pragma unroll
    for (int kb = 0; kb < 2; ++kb) {
      const int k0 = kb * 4 + grp * 2;
      const v2f a = *(const v2f*)&hs[m * HID_ + k0];  // ds_load_b64
      const int n = chunk * 16 + m;                   // all 64 cols valid
      const v2f b = *(const v2f*)&fc2_w[n * HID_ + k0];  // global_load_b64
      acc2 = __builtin_amdgcn_wmma_f32_16x16x4_f32(false, a, false, b,
                                                   (short)0, acc2, false,
                                                   false);
    }
#pragma unroll
    for (int r = 0; r < 8; ++r) {
      const int M = r + grp * 8;
      if (M < B_) {
        const int cg = chunk * 16 + m;
        const float v = acc2[r] + fc2_b[cg];
        ws[GATE_OFF + M * C_ + cg] = 1.f / (1.f + __expf(-v));
      }
    }
  }
}

// ---------------------------------------------------------------------------
// Pass 3: out = x * gate[b,c], float4-vectorized (this pass is the bandwidth
// floor: 302 MB read + 302 MB write).
// ---------------------------------------------------------------------------
__global__ void se3d_scale(const float* __restrict__ x,
                           const float* __restrict__ gate,
                           float* __restrict__ out) {
  const int bc = blockIdx.y;
  const float g = gate[bc];
  const size_t i = (size_t)bc * (DHW_ / 4) +
                   (size_t)blockIdx.x * blockDim.x + threadIdx.x;
  const float4* xp = reinterpret_cast<const float4*>(x);
  float4* op = reinterpret_cast<float4*>(out);
  float4 v = xp[i];
  v.x *= g; v.y *= g; v.z *= g; v.w *= g;
  op[i] = v;
}

extern "C" void kernel_launch(void* const* d_in, const int* in_sizes, int n_in,
                              void* d_out, int out_size, void* d_ws,
                              size_t ws_size, hipStream_t stream) {
  const float* x = (const float*)d_in[0];
  const float* dw_w = (const float*)d_in[1];
  const float* fc1_w = (const float*)d_in[2];
  const float* fc1_b = (const float*)d_in[3];
  const float* fc2_w = (const float*)d_in[4];
  const float* fc2_b = (const float*)d_in[5];
  float* out = (float*)d_out;
  float* ws = (float*)d_ws;

  se3d_init_weff<<<dim3(1), dim3(64), 0, stream>>>(dw_w, ws);
  se3d_conv_pool<<<dim3(B_ * C_ * SLABS), dim3(256), 0, stream>>>(x, ws);
  se3d_mlp_wmma<<<dim3(1), dim3(32), 0, stream>>>(fc1_w, fc1_b, fc2_w, fc2_b,
                                                  ws);
  se3d_scale<<<dim3(DHW_ / (4 * 256), B_ * C_), dim3(256), 0, stream>>>(
      x, ws + GATE_OFF, out);
}